// GradientLoss_16269336117331
// MI455X (gfx1250) — compile-verified
//
#include <hip/hip_runtime.h>
#include <math.h>

// Problem geometry
#define W   512
#define H   512
#define NPL (16 * 3)     // batch * channels planes
#define TX  64           // tile width  (outputs)
#define TY  16           // tile height (outputs)
#define LW  (TX + 2)     // 66: halo width
#define LH  (TY + 2)     // 18: halo height
#define LS  68           // padded LDS row stride (dwords)

typedef __attribute__((ext_vector_type(2))) float v2f;
typedef __attribute__((ext_vector_type(8))) float v8f;

// CDNA5 async global->LDS copy support (probe via __has_builtin; fall back cleanly)
// clang prototype (from diagnostics): (__device__ int*, __shared__ int*, imm offset, imm cpol)
#if defined(__AMDGCN__) && __has_builtin(__builtin_amdgcn_global_load_async_to_lds_b32)
#define HAVE_ASYNC_LDS 1
#define GPTR(p) ((__attribute__((address_space(1))) int*)(p))
#define LPTR(p) ((__attribute__((address_space(3))) int*)(p))
#endif

__device__ __forceinline__ void wait_async_then_barrier() {
#if defined(HAVE_ASYNC_LDS)
#if __has_builtin(__builtin_amdgcn_s_wait_asynccnt)
    __builtin_amdgcn_s_wait_asynccnt(0);
#else
    asm volatile("s_wait_asynccnt 0x0" ::: "memory");
#endif
#endif
    __syncthreads();
}

// |Scharr_x| + |Scharr_y| at (ly,lx) of an LDS tile (center tap is x0 in both kernels)
__device__ __forceinline__ float scharr_mag(const float t[][LS], int ly, int lx) {
    float a = t[ly - 1][lx - 1], b = t[ly - 1][lx], c = t[ly - 1][lx + 1];
    float d = t[ly    ][lx - 1],                    f = t[ly    ][lx + 1];
    float g = t[ly + 1][lx - 1], h = t[ly + 1][lx], i = t[ly + 1][lx + 1];
    float gx = 3.f * (c - a) + 10.f * (f - d) + 3.f * (i - g);
    float gy = 3.f * (g - a) + 10.f * (h - b) + 3.f * (i - c);
    return fabsf(gx) + fabsf(gy);
}

// Stage 1: LDS-tiled Scharr stencil on both tensors, per-block partial sum of
// | (|gx_x|+|gy_x|) - (|gx_g|+|gy_g|) |  ->  partials[blockId]
__global__ __launch_bounds__(256) void scharr_partial_kernel(
    const float* __restrict__ x, const float* __restrict__ gt,
    float* __restrict__ partials)
{
    __shared__ float sx[LH][LS];
    __shared__ float sg[LH][LS];
    __shared__ float wsum[8];

    const int tid = threadIdx.x;
    const int x0  = blockIdx.x * TX;
    const int y0  = blockIdx.y * TY;
    const size_t planeOff = (size_t)blockIdx.z * (size_t)(W * H);
    const float* __restrict__ px = x  + planeOff;
    const float* __restrict__ pg = gt + planeOff;

    // Block-uniform: does the halo stay fully inside the image?
    const bool interior = (x0 > 0) && (x0 + TX < W) && (y0 > 0) && (y0 + TY < H);

#if defined(HAVE_ASYNC_LDS)
    if (interior) {
        // Async DMA halo fill: no VGPR round-trip, tracked by ASYNCcnt.
        for (int i = tid; i < LH * LW; i += 256) {
            int ly = i / LW;
            int lx = i - ly * LW;
            size_t off = (size_t)(y0 + ly - 1) * W + (x0 + lx - 1);
            __builtin_amdgcn_global_load_async_to_lds_b32(GPTR(px + off), LPTR(&sx[ly][lx]), 0, 0);
            __builtin_amdgcn_global_load_async_to_lds_b32(GPTR(pg + off), LPTR(&sg[ly][lx]), 0, 0);
        }
    } else
#endif
    {
        // Guarded scalar halo fill with zero padding (border tiles / fallback)
        for (int i = tid; i < LH * LW; i += 256) {
            int ly = i / LW;
            int lx = i - ly * LW;
            int gy = y0 + ly - 1;
            int gx = x0 + lx - 1;
            bool in = ((unsigned)gy < (unsigned)H) && ((unsigned)gx < (unsigned)W);
            size_t off = (size_t)gy * W + gx;
            sx[ly][lx] = in ? px[off] : 0.f;
            sg[ly][lx] = in ? pg[off] : 0.f;
        }
    }
    wait_async_then_barrier();

    // 256 threads x 4 pixels = 64x16 outputs
    const int ox  = tid & (TX - 1);
    const int oyb = tid >> 6;           // 0..3
    float acc = 0.f;
#pragma unroll
    for (int k = 0; k < 4; ++k) {
        int oy = oyb + 4 * k;           // 0..15
        int ly = oy + 1, lx = ox + 1;
        float rx = scharr_mag(sx, ly, lx);
        float rg = scharr_mag(sg, ly, lx);
        acc += fabsf(rx - rg);
    }

    // wave32 butterfly reduce
#pragma unroll
    for (int off = 16; off > 0; off >>= 1)
        acc += __shfl_down(acc, off, 32);

    if ((tid & 31) == 0) wsum[tid >> 5] = acc;
    __syncthreads();
    if (tid == 0) {
        float s = 0.f;
#pragma unroll
        for (int w = 0; w < 8; ++w) s += wsum[w];
        int bid = blockIdx.x + gridDim.x * (blockIdx.y + gridDim.y * blockIdx.z);
        partials[bid] = s;
    }
}

// Stage 2: single-wave deterministic reduction of block partials using the
// f32 matrix pipe:  C += A(16x4 of partials) x B(4x16 of ones).
// n must be a multiple of 64 (launcher guarantees this) -> branchless b64 loads,
// EXEC stays all-1s as WMMA requires.
__global__ __launch_bounds__(32) void wmma_reduce_kernel(
    const float* __restrict__ partials, int nIter,
    float* __restrict__ out, float scale)
{
    const int lane = threadIdx.x;
    v8f c = {};
    v2f ones = {1.f, 1.f};

    const v2f* __restrict__ p = (const v2f*)partials + lane;  // 32 lanes x v2f = 64 floats/iter
    for (int it = 0; it < nIter; ++it) {
        v2f a = p[(size_t)it * 32];
        c = __builtin_amdgcn_wmma_f32_16x16x4_f32(
            false, a, false, ones, (short)0, c, false, false);
    }

    // Sum this lane's 8 accumulator rows (every C column is identical)
    float s = c[0] + c[1] + c[2] + c[3] + c[4] + c[5] + c[6] + c[7];
    // lanes 0..15 hold sum(rows 0..7), lanes 16..31 hold sum(rows 8..15)
    s += __shfl_xor(s, 16, 32);
    if (lane == 0) out[0] = s * scale;
}

extern "C" void kernel_launch(void* const* d_in, const int* in_sizes, int n_in,
                              void* d_out, int out_size, void* d_ws, size_t ws_size,
                              hipStream_t stream) {
    const float* x  = (const float*)d_in[0];
    const float* gt = (const float*)d_in[1];
    float* out      = (float*)d_out;
    float* partials = (float*)d_ws;   // (512/64)*(512/16)*48 = 12288 floats = 48 KB

    dim3 grid(W / TX, H / TY, NPL);   // 8 x 32 x 48 = 12288 blocks
    scharr_partial_kernel<<<grid, 256, 0, stream>>>(x, gt, partials);

    const int nP = (W / TX) * (H / TY) * NPL;               // 12288 = 192 * 64
    const float scale = 1.0f / ((float)NPL * (float)W * (float)H);
    wmma_reduce_kernel<<<1, 32, 0, stream>>>(partials, nP / 64, out, scale);
}